// Net_89567247991187
// MI455X (gfx1250) — compile-verified
//
#include <hip/hip_runtime.h>
#include <hip/hip_bf16.h>
#include <stdint.h>

// ---------------- problem constants ----------------
#define NG      256
#define NODES   200
#define NNODES  (NG * NODES)   // 51200
#define F_IN    128
#define NHID    64
#define EPG     3200
#define NE      (NG * EPG)     // 819200
#define TRIU    2080
#define NC      10
#define BN_EPS  1e-5f
#define NS_ITERS 30            // Newton-Schulz sign iterations
#define KCHUNK  32             // Gram staging chunk (nodes); 200 = 6*32 + 8

typedef __attribute__((ext_vector_type(2))) float v2f;
typedef __attribute__((ext_vector_type(8))) float v8f;
typedef __attribute__((ext_vector_type(4))) unsigned int v4u;
typedef __attribute__((ext_vector_type(4))) int v4i;
typedef __attribute__((ext_vector_type(8))) int v8i_;

#if __has_builtin(__builtin_amdgcn_tensor_load_to_lds)
#define USE_TDM 1
#else
#define USE_TDM 0
#endif

// f32 WMMA: D(16x16) = A(16x4) * B(4x16) + C   (wave32 f32 path on CDNA5)
__device__ __forceinline__ v8f wmma4(v2f a, v2f b, v8f c) {
  return __builtin_amdgcn_wmma_f32_16x16x4_f32(false, a, false, b, (short)0, c,
                                               false, false);
}

#if USE_TDM
// TDM: 2-D tile (64 floats wide x rows) from global -> LDS, stride 64 floats.
// D# per CDNA5 ISA ch.8: group0 = {count/type/addr}, group1 = {dims/strides}.
__device__ __forceinline__ void tdm_load_tile(const float* gsrc,
                                              unsigned lds_off, int rows) {
  unsigned long long ga = (unsigned long long)(uintptr_t)gsrc;
  v4u g0;
  g0[0] = 1u;                                        // count=1, user mode
  g0[1] = lds_off;                                   // lds_addr (bytes)
  g0[2] = (unsigned)(ga & 0xFFFFFFFFull);            // global_addr[31:0]
  g0[3] = (unsigned)((ga >> 32) & 0x01FFFFFFull)     // global_addr[56:32]
          | (2u << 30);                              // type = 2 ("image")
  v8i_ g1;
  g1[0] = 0x00020000;               // data_size=2 (4B), no multicast/pad/iter
  g1[1] = (int)(64u << 16);         // tensor_dim0 = 64   (bits 79:48, low)
  g1[2] = (int)((unsigned)rows << 16); // tensor_dim1 = rows (bits 111:80, low)
  g1[3] = (int)(64u << 16);         // tile_dim0 = 64     (bits 127:112)
  g1[4] = rows;                     // tile_dim1 = rows; tile_dim2 = 0
  g1[5] = 64;                       // tensor_dim0_stride = 64 (low 32)
  g1[6] = 0;                        // stride0 hi / stride1 lo
  g1[7] = 0;
  v4i z = {};
#if __clang_major__ >= 23
  v8i_ z8 = {};
  __builtin_amdgcn_tensor_load_to_lds(g0, g1, z, z, z8, 0);
#else
  __builtin_amdgcn_tensor_load_to_lds(g0, g1, z, z, 0);
#endif
}
#endif

// ---------------- elementwise helpers ----------------
__global__ void k_fill(float* __restrict__ p, float v, int n) {
  int t = blockIdx.x * blockDim.x + threadIdx.x;
  if (t < n) p[t] = v;
}

__global__ void k_rsqrt_inplace(float* __restrict__ p, int n) {
  int t = blockIdx.x * blockDim.x + threadIdx.x;
  if (t < n) p[t] = rsqrtf(p[t]);
}

__global__ void k_relu_inplace(float* __restrict__ p, int n) {
  int t = blockIdx.x * blockDim.x + threadIdx.x;
  if (t < n) p[t] = fmaxf(p[t], 0.0f);
}

// deg[dst[e]] += 1  (self-loop "+1" comes from the k_fill(1.0) init)
__global__ void k_deg(const int* __restrict__ dst, float* __restrict__ deg, int e) {
  int t = blockIdx.x * blockDim.x + threadIdx.x;
  if (t < e) unsafeAtomicAdd(&deg[dst[t]], 1.0f);
}

__global__ void k_norm(float* __restrict__ nrm, const int* __restrict__ src,
                       const int* __restrict__ dst, const float* __restrict__ dinv,
                       int e) {
  int t = blockIdx.x * blockDim.x + threadIdx.x;
  if (t < e) nrm[t] = dinv[src[t]] * dinv[dst[t]];
}

// agg[i][0:4] = bias + hw[i][0:4] * dinv[i]^2   (self-loop + bias init), float4
__global__ void k_agg_init(float* __restrict__ agg, const float* __restrict__ hw,
                           const float* __restrict__ dinv,
                           const float* __restrict__ bias, int n /* NNODES*16 */) {
  int t = blockIdx.x * blockDim.x + threadIdx.x;
  if (t < n) {
    int i = t >> 4, f = (t & 15) << 2;
    float di = dinv[i];
    float s = di * di;
    float4 v = *(const float4*)(hw + (size_t)i * NHID + f);
    const float4 b = *(const float4*)(bias + f);
    float4 o;
    o.x = b.x + v.x * s; o.y = b.y + v.y * s;
    o.z = b.z + v.z * s; o.w = b.w + v.w * s;
    *(float4*)(agg + (size_t)i * NHID + f) = o;
  }
}

// scatter-add of normalized messages, 4 features per thread
__global__ void k_agg_edges(float* __restrict__ agg, const float* __restrict__ hw,
                            const int* __restrict__ src, const int* __restrict__ dst,
                            const float* __restrict__ nrm, int n /* NE*16 */) {
  int t = blockIdx.x * blockDim.x + threadIdx.x;
  if (t < n) {
    int e = t >> 4, f = (t & 15) << 2;
    int s = src[e], d = dst[e];
    float w = nrm[e];
    float4 v = *(const float4*)(hw + (size_t)s * NHID + f);
    float* o = agg + (size_t)d * NHID + f;
    unsafeAtomicAdd(o + 0, v.x * w);
    unsafeAtomicAdd(o + 1, v.y * w);
    unsafeAtomicAdd(o + 2, v.z * w);
    unsafeAtomicAdd(o + 3, v.w * w);
  }
}

// ---------------- generic row-major WMMA f32 GEMM ----------------
// C[M x N] = A[M x K] @ B[K x N] (+bias) (+relu).
// Each wave owns NT consecutive 16x16 n-tiles of one 16-row m-stripe, so the
// A operand is fetched once per K-step and reused NT times (NT accumulators).
// K % 4 == 0, M % 16 == 0; N arbitrary (uniform-branch tail, branchless mask).
template <int NT>
__global__ void k_gemm(const float* __restrict__ A, int lda,
                       const float* __restrict__ B, int ldb,
                       float* __restrict__ C, int ldc,
                       int M, int N, int K,
                       const float* __restrict__ bias, int flags) {
  int tN = (N + 15) >> 4;
  int nG = (tN + NT - 1) / NT;
  int wid = blockIdx.x * (blockDim.x >> 5) + (threadIdx.x >> 5);
  int total = (M >> 4) * nG;
  if (wid >= total) return;                  // whole-wave exit, EXEC stays full
  int lane = threadIdx.x & 31;
  int m0 = (wid / nG) << 4;
  int n0 = (wid % nG) * (NT << 4);
  int r = lane & 15, h = lane >> 4;
  const float* Arow = A + (size_t)(m0 + r) * lda;

  v8f acc[NT];
#pragma unroll
  for (int q = 0; q < NT; ++q) acc[q] = (v8f){};

  if (n0 + (NT << 4) <= N) {
    // full tiles: no masking anywhere in the hot loop
    const float* Bc = B + n0 + r;
    for (int k = 0; k < K; k += 4) {
      int kk = k + (h << 1);
      v2f a = *(const v2f*)(Arow + kk);
      __builtin_prefetch(Arow + kk + 32, 0, 1);   // global_prefetch_b8
      const float* B0 = Bc + (size_t)kk * ldb;
      const float* B1 = B0 + ldb;
#pragma unroll
      for (int q = 0; q < NT; ++q) {
        v2f b; b.x = B0[q << 4]; b.y = B1[q << 4];
        acc[q] = wmma4(a, b, acc[q]);
      }
    }
  } else {
    // tail group (N % (NT*16) != 0): clamp + multiply mask, no EXEC churn
    int nc[NT]; float msk[NT];
#pragma unroll
    for (int q = 0; q < NT; ++q) {
      int n = n0 + (q << 4) + r;
      nc[q] = n < N ? n : N - 1;
      msk[q] = n < N ? 1.0f : 0.0f;
    }
    for (int k = 0; k < K; k += 4) {
      int kk = k + (h << 1);
      v2f a = *(const v2f*)(Arow + kk);
#pragma unroll
      for (int q = 0; q < NT; ++q) {
        v2f b;
        b.x = B[(size_t)kk * ldb + nc[q]] * msk[q];
        b.y = B[(size_t)(kk + 1) * ldb + nc[q]] * msk[q];
        acc[q] = wmma4(a, b, acc[q]);
      }
    }
  }

#pragma unroll
  for (int q = 0; q < NT; ++q) {
    int n = n0 + (q << 4) + r;
    if (n < N) {
      float bv = bias ? bias[n] : 0.0f;
#pragma unroll
      for (int rr = 0; rr < 8; ++rr) {
        int m = m0 + rr + (h << 3);
        float v = acc[q][rr] + bv;
        if (flags & 1) v = fmaxf(v, 0.0f);
        C[(size_t)m * ldc + n] = v;
      }
    }
  }
}

// ---------------- Grassmann pooling: projector via matrix sign ----------------
// Gram accumulation over one LDS-resident chunk of CNT nodes (compile-time
// trip counts so DS loads pipeline ahead of the WMMA group).
template <int CNT>
__device__ __forceinline__ void gram_accum(const float* Ash, v8f acc[4],
                                           int i0, int r, int h) {
#pragma unroll
  for (int q = 0; q < 4; ++q) {
    int j0 = q << 4;
#pragma unroll
    for (int k = 0; k < CNT; k += 4) {
      int kk = k + (h << 1);
      v2f a, b;
      a.x = Ash[kk * NHID + i0 + r];
      a.y = Ash[(kk + 1) * NHID + i0 + r];
      b.x = Ash[kk * NHID + j0 + r];
      b.y = Ash[(kk + 1) * NHID + j0 + r];
      acc[q] = wmma4(a, b, acc[q]);
    }
  }
}

// Z = alpha*(X@Y) + addScale*addM, all 64x64 row-major in LDS. 4 waves, each
// owns a 16-row stripe (4 tiles). WMMA operands come from LDS (ds_load).
__device__ __forceinline__ void lds_mm(const float* X, const float* Y, float* Z,
                                       const float* addM, float alpha,
                                       float addScale, int wave, int lane) {
  int r = lane & 15, h = lane >> 4;
  int i0 = wave << 4;
  const float* Xr = X + (i0 + r) * NHID;
  for (int q = 0; q < 4; ++q) {
    int j0 = q << 4;
    v8f c = {};
#pragma unroll 4
    for (int k = 0; k < NHID; k += 4) {
      int kk = k + (h << 1);
      v2f a = *(const v2f*)(Xr + kk);
      v2f b;
      b.x = Y[kk * NHID + j0 + r];
      b.y = Y[(kk + 1) * NHID + j0 + r];
      c = wmma4(a, b, c);
    }
    for (int rr = 0; rr < 8; ++rr) {
      int m = i0 + rr + (h << 3), nn = j0 + r;
      float v = alpha * c[rr];
      if (addM) v += addScale * addM[m * NHID + nn];
      Z[m * NHID + nn] = v;
    }
  }
}

// One block (128 thr = 4 waves) per graph.
// G = A A^T (A = H2^T, 64x200) built from LDS-staged chunks (TDM when
// available); S0 = (G-0.25I)/||.||F; Newton-Schulz S <- 1.5S - 0.5 S^3 ==>
// sign(G - 0.25 I); P = (I+sign)/2 == U[:,S>0.5] U[:,S>0.5]^T; emit triu.
// LDS: 8KB stage + 3*16KB iterate + 32B = ~56.3KB.
__global__ void k_grasspool(const float* __restrict__ H2, float* __restrict__ P) {
  __shared__ float Ash[KCHUNK * NHID];   // 8 KB staging chunk
  __shared__ float S[NHID * NHID];
  __shared__ float T[NHID * NHID];
  __shared__ float U[NHID * NHID];
  __shared__ float red[8];
  int g = blockIdx.x;
  int tid = threadIdx.x;
  int wave = tid >> 5, lane = tid & 31;
  int r = lane & 15, h = lane >> 4;
  const float* gsrc = H2 + (size_t)g * NODES * NHID;

  // Gram: G[i][j] = sum_n A[n][i] * A[n][j], accumulated over staged chunks
  int i0 = wave << 4;
  v8f acc[4];
#pragma unroll
  for (int q = 0; q < 4; ++q) acc[q] = (v8f){};

#if USE_TDM
  unsigned ashOff = (unsigned)(uintptr_t)(void*)Ash;
  for (int kb = 0; kb < NODES - KCHUNK + 1; kb += KCHUNK) {   // 6 full chunks
    if (wave == 0) {
      tdm_load_tile(gsrc + (size_t)kb * NHID, ashOff, KCHUNK);
      __builtin_amdgcn_s_wait_tensorcnt(0);
    }
    __syncthreads();
    gram_accum<KCHUNK>(Ash, acc, i0, r, h);
    __syncthreads();
  }
  if (wave == 0) {                                            // 8-row tail
    tdm_load_tile(gsrc + (size_t)(NODES - 8) * NHID, ashOff, 8);
    __builtin_amdgcn_s_wait_tensorcnt(0);
  }
  __syncthreads();
  gram_accum<8>(Ash, acc, i0, r, h);
  __syncthreads();
#else
  const float4* src4 = (const float4*)gsrc;
  for (int kb = 0; kb < NODES - KCHUNK + 1; kb += KCHUNK) {
    for (int i = tid; i < KCHUNK * (NHID / 4); i += 128)
      ((float4*)Ash)[i] = src4[kb * (NHID / 4) + i];
    __syncthreads();
    gram_accum<KCHUNK>(Ash, acc, i0, r, h);
    __syncthreads();
  }
  for (int i = tid; i < 8 * (NHID / 4); i += 128)
    ((float4*)Ash)[i] = src4[(NODES - 8) * (NHID / 4) + i];
  __syncthreads();
  gram_accum<8>(Ash, acc, i0, r, h);
  __syncthreads();
#endif

#pragma unroll
  for (int q = 0; q < 4; ++q) {
    int j0 = q << 4;
    for (int rr = 0; rr < 8; ++rr) {
      int m = i0 + rr + (h << 3), nn = j0 + r;
      S[m * NHID + nn] = acc[q][rr] - (m == nn ? 0.25f : 0.0f);
    }
  }
  __syncthreads();

  // Frobenius-norm scale (spectral radius <= 1 guarantees NS convergence)
  float s = 0.0f;
  for (int idx = tid; idx < NHID * NHID; idx += 128) { float v = S[idx]; s += v * v; }
  for (int off = 16; off; off >>= 1) s += __shfl_down(s, off, 32);
  if (lane == 0) red[wave] = s;
  __syncthreads();
  if (tid == 0) red[4] = rsqrtf(red[0] + red[1] + red[2] + red[3] + 1e-12f);
  __syncthreads();
  float inv = red[4];
  for (int idx = tid; idx < NHID * NHID; idx += 128) S[idx] *= inv;
  __syncthreads();

  float* Sp = S;
  float* Up = U;
  for (int it = 0; it < NS_ITERS; ++it) {
    lds_mm(Sp, Sp, T, nullptr, 1.0f, 0.0f, wave, lane);   // T = S^2
    __syncthreads();
    lds_mm(Sp, T, Up, Sp, -0.5f, 1.5f, wave, lane);       // S' = 1.5S - 0.5 S*T
    __syncthreads();
    float* tmp = Sp; Sp = Up; Up = tmp;
  }

  // Projector upper triangle -> P[g, :]
  float* out = P + (size_t)g * TRIU;
  int base = 0;
  for (int i = 0; i < NHID; ++i) {
    for (int j = i + tid; j < NHID; j += 128)
      out[base + (j - i)] = 0.5f * Sp[i * NHID + j] + (i == j ? 0.5f : 0.0f);
    base += NHID - i;
  }
}

// ---------------- BatchNorm over the batch dim (training stats) ----------------
// One block per column, 256 rows. y = g*(x-m)*rsqrt(var+eps)+b (+relu)
__global__ void k_bn(const float* __restrict__ X, float* __restrict__ Y,
                     const float* __restrict__ gam, const float* __restrict__ bet,
                     int C, int relu) {
  int c = blockIdx.x, t = threadIdx.x;
  float x = X[(size_t)t * C + c];
  float s = x, s2 = x * x;
  for (int off = 16; off; off >>= 1) {
    s  += __shfl_down(s,  off, 32);
    s2 += __shfl_down(s2, off, 32);
  }
  __shared__ float sh[16];
  int wave = t >> 5, lane = t & 31;
  if (lane == 0) { sh[wave] = s; sh[8 + wave] = s2; }
  __syncthreads();
  if (t == 0) {
    float ts = 0.0f, t2 = 0.0f;
    for (int w = 0; w < 8; ++w) { ts += sh[w]; t2 += sh[8 + w]; }
    float m = ts * (1.0f / NG);
    float v = t2 * (1.0f / NG) - m * m;
    sh[0] = m; sh[1] = rsqrtf(v + BN_EPS);
  }
  __syncthreads();
  float y = gam[c] * (x - sh[0]) * sh[1] + bet[c];
  if (relu) y = fmaxf(y, 0.0f);
  Y[(size_t)t * C + c] = y;
}

// ---------------- launcher ----------------
static inline int cdiv(int a, int b) { return (a + b - 1) / b; }

extern "C" void kernel_launch(void* const* d_in, const int* in_sizes, int n_in,
                              void* d_out, int out_size, void* d_ws, size_t ws_size,
                              hipStream_t stream) {
  (void)in_sizes; (void)n_in; (void)out_size; (void)ws_size;
  const float* x   = (const float*)d_in[0];
  const int*   ei  = (const int*)d_in[1];
  const int*   src = ei;
  const int*   dst = ei + NE;
  // d_in[2] = batch (implicit from fixed layout)
  const float* W1  = (const float*)d_in[3];
  const float* b1  = (const float*)d_in[4];
  const float* W2  = (const float*)d_in[5];
  const float* b2  = (const float*)d_in[6];
  const float* Wf1 = (const float*)d_in[7];
  const float* bf1 = (const float*)d_in[8];
  const float* g1  = (const float*)d_in[9];
  const float* be1 = (const float*)d_in[10];
  const float* Wf2 = (const float*)d_in[11];
  const float* bf2 = (const float*)d_in[12];
  const float* g2  = (const float*)d_in[13];
  const float* be2 = (const float*)d_in[14];
  const float* Wf3 = (const float*)d_in[15];
  const float* bf3 = (const float*)d_in[16];
  const float* g3  = (const float*)d_in[17];
  const float* be3 = (const float*)d_in[18];

  // workspace layout
  char* w = (char*)d_ws;
  auto take = [&](size_t nflt) {
    float* p = (float*)w; w += ((nflt * 4 + 255) & ~255ull); return p;
  };
  float* dinv  = take(NNODES);
  float* nrm   = take(NE);
  float* bufA  = take((size_t)NNODES * NHID);
  float* bufB  = take((size_t)NNODES * NHID);
  float* Pfeat = take((size_t)NG * TRIU);
  float* Z1p   = take((size_t)NG * 1024);
  float* Z1    = take((size_t)NG * 1024);
  float* Z2p   = take((size_t)NG * 512);
  float* Z2    = take((size_t)NG * 512);
  float* Z3p   = take((size_t)NG * NC);

  auto gemm1 = [&](const float* A, int lda, const float* B, int ldb, float* C,
                   int ldc, int M, int N, int K, const float* bias, int flags) {
    int tN = (N + 15) >> 4;
    int waves = (M >> 4) * tN;
    int blocks = (waves + 7) >> 3;
    k_gemm<1><<<blocks, 256, 0, stream>>>(A, lda, B, ldb, C, ldc, M, N, K, bias, flags);
  };
  auto gemm4 = [&](const float* A, int lda, const float* B, int ldb, float* C,
                   int ldc, int M, int N, int K, const float* bias, int flags) {
    int tN = (N + 15) >> 4;
    int waves = (M >> 4) * ((tN + 3) >> 2);
    int blocks = (waves + 7) >> 3;
    k_gemm<4><<<blocks, 256, 0, stream>>>(A, lda, B, ldb, C, ldc, M, N, K, bias, flags);
  };

  // degrees (self-loop via init=1), dinv, per-edge norm
  k_fill<<<cdiv(NNODES, 256), 256, 0, stream>>>(dinv, 1.0f, NNODES);
  k_deg<<<cdiv(NE, 256), 256, 0, stream>>>(dst, dinv, NE);
  k_rsqrt_inplace<<<cdiv(NNODES, 256), 256, 0, stream>>>(dinv, NNODES);
  k_norm<<<cdiv(NE, 256), 256, 0, stream>>>(nrm, src, dst, dinv, NE);

  // GCN layer 1: HW = x@W1 (A reused across all 4 n-tiles); agg; relu
  gemm4(x, F_IN, W1, NHID, bufA, NHID, NNODES, NHID, F_IN, nullptr, 0);
  k_agg_init<<<cdiv(NNODES * 16, 256), 256, 0, stream>>>(bufB, bufA, dinv, b1, NNODES * 16);
  k_agg_edges<<<cdiv(NE * 16, 256), 256, 0, stream>>>(bufB, bufA, src, dst, nrm, NE * 16);
  k_relu_inplace<<<cdiv(NNODES * NHID, 256), 256, 0, stream>>>(bufB, NNODES * NHID);

  // GCN layer 2
  gemm4(bufB, NHID, W2, NHID, bufA, NHID, NNODES, NHID, NHID, nullptr, 0);
  k_agg_init<<<cdiv(NNODES * 16, 256), 256, 0, stream>>>(bufB, bufA, dinv, b2, NNODES * 16);
  k_agg_edges<<<cdiv(NE * 16, 256), 256, 0, stream>>>(bufB, bufA, src, dst, nrm, NE * 16);
  k_relu_inplace<<<cdiv(NNODES * NHID, 256), 256, 0, stream>>>(bufB, NNODES * NHID);

  // Grassmann pooling: one workgroup per graph, WMMA sign-iteration in LDS
  k_grasspool<<<NG, 128, 0, stream>>>(bufB, Pfeat);

  // FC stack with batch-stat BN
  gemm1(Pfeat, TRIU, Wf1, 1024, Z1p, 1024, NG, 1024, TRIU, bf1, 0);
  k_bn<<<1024, NG, 0, stream>>>(Z1p, Z1, g1, be1, 1024, 1);
  gemm1(Z1, 1024, Wf2, 512, Z2p, 512, NG, 512, 1024, bf2, 0);
  k_bn<<<512, NG, 0, stream>>>(Z2p, Z2, g2, be2, 512, 1);
  gemm1(Z2, 512, Wf3, NC, Z3p, NC, NG, NC, 512, bf3, 0);
  k_bn<<<NC, NG, 0, stream>>>(Z3p, (float*)d_out, g3, be3, NC, 0);
}